// NewSpike_72318659330517
// MI455X (gfx1250) — compile-verified
//
#include <hip/hip_runtime.h>

// CDNA5 / gfx1250: wave32, WMMA 16x16x4 F32.
typedef __attribute__((ext_vector_type(2))) float v2f;
typedef __attribute__((ext_vector_type(8))) float v8f;

#define THETA_F     0.799f          // 16/20 - 0.001
#define INV_THETA_F (1.0f / 0.799f)
#define INV_T_F     (1.0f / 16.0f)

// One wave handles one 16-pixel x 16-timestep tile (1 KB in, 1 KB out).
// Reduction over T done by V_WMMA_F32_16X16X4_F32 against an all-ones B:
//   D[m][n] = sum_t x[m][t]   (replicated over columns n)
// Lane's column index n doubles as the timestep this lane emits, via the
// closed form  S_i = min(i, floor(i*u/theta)),  spike_t = S_{t+1} - S_t.
__global__ __launch_bounds__(256) void
NewSpike_72318659330517_kernel(const float* __restrict__ x,
                               float* __restrict__ out,
                               int ntiles) {
  const int lane = threadIdx.x & 31;
  const int wid  = blockIdx.x * 8 + (threadIdx.x >> 5);   // 8 waves / block
  if (wid >= ntiles) return;                               // wave-uniform guard

  const int row  = lane & 15;   // A-matrix row M; also this lane's N column / timestep t
  const int half = lane >> 4;   // selects K{0,1} vs K{2,3} within each K=4 slice

  // A-matrix 16x4 f32 layout: lanes 0-15 hold K=0,1 ; lanes 16-31 hold K=2,3.
  // Slice c covers K = 4c..4c+3  ->  lane needs floats [row*16 + 4c + 2*half .. +1]
  const float* src = x + (size_t)wid * 256 + row * 16 + half * 2;
  v2f a0 = *(const v2f*)(src + 0);
  v2f a1 = *(const v2f*)(src + 4);
  v2f a2 = *(const v2f*)(src + 8);
  v2f a3 = *(const v2f*)(src + 12);

  v2f b;  b.x = 1.0f; b.y = 1.0f;      // ones matrix: layout-independent
  v8f acc = {};
  acc = __builtin_amdgcn_wmma_f32_16x16x4_f32(false, a0, false, b, (short)0, acc, false, false);
  acc = __builtin_amdgcn_wmma_f32_16x16x4_f32(false, a1, false, b, (short)0, acc, false, false);
  acc = __builtin_amdgcn_wmma_f32_16x16x4_f32(false, a2, false, b, (short)0, acc, false, false);
  acc = __builtin_amdgcn_wmma_f32_16x16x4_f32(false, a3, false, b, (short)0, acc, false, false);

  // C/D layout: VGPR j holds row M = j + 8*half, column N = lane&15 (= timestep t).
  const float ft  = (float)row;        // t
  const float ft1 = ft + 1.0f;         // t+1

  float* dst = out + (size_t)wid * 256 + (size_t)(half * 8) * 16 + row;
#pragma unroll
  for (int j = 0; j < 8; ++j) {
    float u  = acc[j] * INV_T_F;                 // mean drive
    float p  = u * INV_THETA_F;                  // u / theta
    float s1 = fminf(ft1, floorf(ft1 * p));      // S_{t+1}
    float s0 = fminf(ft,  floorf(ft  * p));      // S_t
    dst[j * 16] = s1 - s0;                       // spike at timestep t, row j+8*half
  }
}

extern "C" void kernel_launch(void* const* d_in, const int* in_sizes, int n_in,
                              void* d_out, int out_size, void* d_ws, size_t ws_size,
                              hipStream_t stream) {
  (void)n_in; (void)d_ws; (void)ws_size; (void)out_size;
  const float* x   = (const float*)d_in[0];
  float*       out = (float*)d_out;

  const long long n      = (long long)in_sizes[0];   // 32*128*32*32*16 = 67,108,864
  const int       ntiles = (int)(n / 256);           // 16 pixels x 16 T per tile
  const int       blocks = (ntiles + 7) / 8;         // 8 waves per 256-thread block

  NewSpike_72318659330517_kernel<<<blocks, 256, 0, stream>>>(x, out, ntiles);
}